// DynamicSelectiveHyperNet_41128606826973
// MI455X (gfx1250) — compile-verified
//
#include <hip/hip_runtime.h>
#include <hip/hip_bf16.h>

typedef float v2f __attribute__((ext_vector_type(2)));
typedef float v8f __attribute__((ext_vector_type(8)));

#define T_DIM 101770
// ws layout (floats)
#define WS_HIN  0                      // [32][96]   (pb = p*8+b major)
#define WS_HMID (32 * 96)              // [8][32][32]
#define WS_GATE (32 * 96 + 8 * 32 * 32) // [8][8] row-major (row index used as 'h' at consume time)

static __device__ __forceinline__ v8f wmma_f32_k4(v2f a, v2f b, v8f c) {
  // D(16x16,f32) = A(16x4,f32) * B(4x16,f32) + C
  return __builtin_amdgcn_wmma_f32_16x16x4_f32(false, a, false, b, (short)0, c,
                                               false, false);
}

// ---------------------------------------------------------------------------
// Kernel 1: tiny preamble on one workgroup.
// feats, gate(softmax), hin = [feats|embed], hmid = relu(hin @ genW1^T + b)
// ---------------------------------------------------------------------------
__global__ __launch_bounds__(1024) void hyper_prep(
    const float* __restrict__ x,      const float* __restrict__ feW1,
    const float* __restrict__ feb1,   const float* __restrict__ feW2,
    const float* __restrict__ feb2,   const float* __restrict__ embeds,
    const float* __restrict__ genW1,  const float* __restrict__ genb1,
    const float* __restrict__ gateW,  const float* __restrict__ gateb,
    float* __restrict__ ws) {
  __shared__ float s_h1[8][128];
  __shared__ float s_feats[8][64];
  __shared__ float s_hin[32][96];
  __shared__ float s_logit[8][8];
  const int tid = threadIdx.x;

  // h1 = relu(x @ feW1^T + feb1) : [8][128]
  for (int item = tid; item < 8 * 128; item += 1024) {
    const int bb = item >> 7, j = item & 127;
    float acc = feb1[j];
    const float* xr = x + bb * 784;
    const float* wr = feW1 + j * 784;
    for (int i = 0; i < 784; ++i) acc += xr[i] * wr[i];
    s_h1[bb][j] = fmaxf(acc, 0.0f);
  }
  __syncthreads();

  // feats = h1 @ feW2^T + feb2 : [8][64]
  for (int item = tid; item < 8 * 64; item += 1024) {
    const int bb = item >> 6, f = item & 63;
    float acc = feb2[f];
    const float* wr = feW2 + f * 128;
    for (int j = 0; j < 128; ++j) acc += s_h1[bb][j] * wr[j];
    s_feats[bb][f] = acc;
  }
  __syncthreads();

  // gate logits : [8][8]
  if (tid < 64) {
    const int bb = tid >> 3, hh = tid & 7;
    float acc = gateb[hh];
    const float* wr = gateW + hh * 64;
    for (int f = 0; f < 64; ++f) acc += s_feats[bb][f] * wr[f];
    s_logit[bb][hh] = acc;
  }
  __syncthreads();

  // softmax over axis=1 per row; store row-major to ws
  if (tid < 8) {
    const int bb = tid;
    float m = -1e30f;
    for (int hh = 0; hh < 8; ++hh) m = fmaxf(m, s_logit[bb][hh]);
    float e[8], sum = 0.0f;
    for (int hh = 0; hh < 8; ++hh) { e[hh] = __expf(s_logit[bb][hh] - m); sum += e[hh]; }
    const float inv = 1.0f / sum;
    for (int hh = 0; hh < 8; ++hh) ws[WS_GATE + bb * 8 + hh] = e[hh] * inv;
  }

  // hin[pb][i] = concat(feats[b], embeds[p]) : [32][96]
  for (int item = tid; item < 32 * 96; item += 1024) {
    const int pb = item / 96, i = item % 96;
    const int p = pb >> 3, bb = pb & 7;
    const float v = (i < 64) ? s_feats[bb][i] : embeds[p * 32 + (i - 64)];
    s_hin[pb][i] = v;
    ws[WS_HIN + item] = v;
  }
  __syncthreads();

  // hmid[h][pb][k] = relu(hin[pb] . genW1[h][k] + genb1[h][k]) : [8][32][32]
  for (int item = tid; item < 8 * 32 * 32; item += 1024) {
    const int h = item >> 10, rem = item & 1023;
    const int pb = rem >> 5, k = rem & 31;
    float acc = genb1[h * 32 + k];
    const float* wr = genW1 + (size_t)(h * 32 + k) * 96;
    for (int i = 0; i < 96; ++i) acc += s_hin[pb][i] * wr[i];
    ws[WS_HMID + item] = fmaxf(acc, 0.0f);
  }
}

// ---------------------------------------------------------------------------
// Kernel 2: memory-bound stream over T. Each wave owns 16 t-columns.
// Per head h: D(32x16) = hmid[h](32x32) @ genW2[h,t,:]^T  (f32 WMMA, K=4 x8)
//             E(32x16) = hin(32x96)     @ attW[h,t,:]^T   (f32 WMMA, K=4 x24)
// combined += (D + bg) * sigmoid(E + ba) * gate[h][b]
// ---------------------------------------------------------------------------
__global__ __launch_bounds__(256) void hyper_main(
    const float* __restrict__ genW2, const float* __restrict__ genb2,
    const float* __restrict__ attW,  const float* __restrict__ attb,
    const float* __restrict__ ws,    float* __restrict__ out) {
  const int lane = threadIdx.x & 31;
  const int wave = threadIdx.x >> 5;
  const int tIdx = blockIdx.x * 8 + wave;
  const int numTiles = (T_DIM + 15) >> 4;
  if (tIdx >= numTiles) return;  // whole-wave uniform exit

  const int n  = lane & 15;   // N index (t within tile) / A-row (M%16)
  const int hi = lane >> 4;   // selects K-pair 2*hi within each K=4 chunk
  const int t  = tIdx * 16 + n;
  const int tcl = (t < T_DIM) ? t : (T_DIM - 1);

  const float* hmBase  = ws + WS_HMID + n * 32 + 2 * hi;  // hmid A frags
  const float* hinBase = ws + WS_HIN  + n * 96 + 2 * hi;  // hin  A frags

  v8f comb0 = {0.f, 0.f, 0.f, 0.f, 0.f, 0.f, 0.f, 0.f};
  v8f comb1 = {0.f, 0.f, 0.f, 0.f, 0.f, 0.f, 0.f, 0.f};

#pragma unroll 1
  for (int h = 0; h < 8; ++h) {
    v8f aw0 = {0.f, 0.f, 0.f, 0.f, 0.f, 0.f, 0.f, 0.f};
    v8f aw1 = {0.f, 0.f, 0.f, 0.f, 0.f, 0.f, 0.f, 0.f};
    v8f aa0 = {0.f, 0.f, 0.f, 0.f, 0.f, 0.f, 0.f, 0.f};
    v8f aa1 = {0.f, 0.f, 0.f, 0.f, 0.f, 0.f, 0.f, 0.f};

    const float* gB  = genW2 + ((size_t)h * T_DIM + (size_t)tcl) * 32 + 2 * hi;
    const float* aB  = attW  + ((size_t)h * T_DIM + (size_t)tcl) * 96 + 2 * hi;
    const float* hmA = hmBase + h * (32 * 32);

    // gen path: K = 32 -> 8 chunks of K=4; two M-tiles (pb 0-15, 16-31)
#pragma unroll
    for (int kc = 0; kc < 8; ++kc) {
      const v2f b  = __builtin_nontemporal_load((const v2f*)(gB + 4 * kc));
      const v2f a0 = *(const v2f*)(hmA + 4 * kc);
      const v2f a1 = *(const v2f*)(hmA + 16 * 32 + 4 * kc);
      aw0 = wmma_f32_k4(a0, b, aw0);
      aw1 = wmma_f32_k4(a1, b, aw1);
    }
    // att path: K = 96 -> 24 chunks of K=4
#pragma unroll
    for (int kc = 0; kc < 24; ++kc) {
      const v2f b  = __builtin_nontemporal_load((const v2f*)(aB + 4 * kc));
      const v2f a0 = *(const v2f*)(hinBase + 4 * kc);
      const v2f a1 = *(const v2f*)(hinBase + 16 * 96 + 4 * kc);
      aa0 = wmma_f32_k4(a0, b, aa0);
      aa1 = wmma_f32_k4(a1, b, aa1);
    }

    const float bg = __builtin_nontemporal_load(genb2 + (size_t)h * T_DIM + tcl);
    const float ba = __builtin_nontemporal_load(attb  + (size_t)h * T_DIM + tcl);

    // D layout: lane L, reg r -> M = r + 8*(L/16), N = L%16  => b = r, p = 2*tile + hi
#pragma unroll
    for (int r = 0; r < 8; ++r) {
      const float g  = ws[WS_GATE + h * 8 + r];  // uniform -> scalar load
      const float w0 = aw0[r] + bg;
      const float s0 = 1.0f / (1.0f + __expf(-(aa0[r] + ba)));
      comb0[r] += w0 * s0 * g;
      const float w1 = aw1[r] + bg;
      const float s1 = 1.0f / (1.0f + __expf(-(aa1[r] + ba)));
      comb1[r] += w1 * s1 * g;
    }
  }

  if (t < T_DIM) {
    const int p0 = hi;      // M-tile 0 -> p = 0 + hi
    const int p1 = 2 + hi;  // M-tile 1 -> p = 2 + hi
#pragma unroll
    for (int r = 0; r < 8; ++r) {
      out[(size_t)r * (4 * (size_t)T_DIM) + (size_t)p0 * T_DIM + t] = comb0[r];
      out[(size_t)r * (4 * (size_t)T_DIM) + (size_t)p1 * T_DIM + t] = comb1[r];
    }
  }
}

extern "C" void kernel_launch(void* const* d_in, const int* in_sizes, int n_in,
                              void* d_out, int out_size, void* d_ws, size_t ws_size,
                              hipStream_t stream) {
  (void)in_sizes; (void)n_in; (void)out_size; (void)ws_size;
  const float* x      = (const float*)d_in[0];
  const float* feW1   = (const float*)d_in[1];
  const float* feb1   = (const float*)d_in[2];
  const float* feW2   = (const float*)d_in[3];
  const float* feb2   = (const float*)d_in[4];
  const float* embeds = (const float*)d_in[5];
  const float* genW1  = (const float*)d_in[6];
  const float* genb1  = (const float*)d_in[7];
  const float* genW2  = (const float*)d_in[8];
  const float* genb2  = (const float*)d_in[9];
  const float* attW   = (const float*)d_in[10];
  const float* attb   = (const float*)d_in[11];
  const float* gateW  = (const float*)d_in[12];
  const float* gateb  = (const float*)d_in[13];
  float* out = (float*)d_out;
  float* ws  = (float*)d_ws;

  hyper_prep<<<1, 1024, 0, stream>>>(x, feW1, feb1, feW2, feb2, embeds,
                                     genW1, genb1, gateW, gateb, ws);

  const int numTiles = (T_DIM + 15) / 16;      // 6361 wave-tiles of 16 t
  const int blocks   = (numTiles + 7) / 8;     // 8 waves per 256-thread block
  hyper_main<<<blocks, 256, 0, stream>>>(genW2, genb2, attW, attb, ws, out);
}